// FlashHead_1975684956889
// MI455X (gfx1250) — compile-verified
//
#include <hip/hip_runtime.h>

#define D_DIM  4096
#define NC     2048
#define CS     64
#define NPROBE 128
#define NCAND  (NPROBE * CS)   // 8192

typedef __attribute__((ext_vector_type(2))) float v2f;
typedef __attribute__((ext_vector_type(8))) float v8f;

// ---------------- Phase 1a: partial dot(h, centroid_col) and partial norm^2 ----
// grid (NC/256, 16), block 256.  Lanes -> clusters (coalesced), split over D.
__global__ void k_sims_partial(const float* __restrict__ h,
                               const float* __restrict__ cent,
                               float* __restrict__ pdot,
                               float* __restrict__ pnrm) {
  const int c  = blockIdx.x * 256 + threadIdx.x;
  const int d0 = blockIdx.y * (D_DIM / 16);
  const float* p = cent + (size_t)d0 * NC + c;
  float dot = 0.f, nrm = 0.f;
  #pragma unroll 8
  for (int d = 0; d < D_DIM / 16; ++d) {
    const float v  = p[(size_t)d * NC];
    const float hv = h[d0 + d];          // uniform -> scalar load
    dot = fmaf(v, hv, dot);
    nrm = fmaf(v, v, nrm);
  }
  pdot[blockIdx.y * NC + c] = dot;
  pnrm[blockIdx.y * NC + c] = nrm;
}

// ---------------- Phase 1b: reduce partials -> sims[c] ----------------
__global__ void k_sims_reduce(const float* __restrict__ pdot,
                              const float* __restrict__ pnrm,
                              float* __restrict__ sims) {
  const int c = blockIdx.x * 256 + threadIdx.x;
  float dot = 0.f, nrm = 0.f;
  #pragma unroll
  for (int y = 0; y < 16; ++y) { dot += pdot[y * NC + c]; nrm += pnrm[y * NC + c]; }
  sims[c] = dot / sqrtf(nrm);
}

// ---------------- Phase 2: exact, sorted top-128 by rank counting --------------
// grid NC/256, block 256.  Unique writer per rank -> deterministic.
__global__ void k_topk(const float* __restrict__ sims, int* __restrict__ top) {
  __shared__ float s[NC];
  for (int i = threadIdx.x; i < NC; i += 256) s[i] = sims[i];
  __syncthreads();
  const int   i    = blockIdx.x * 256 + threadIdx.x;
  const float mine = s[i];
  int rank = 0;
  for (int j0 = 0; j0 < NC; j0 += 4) {
    const float4 o = *(const float4*)&s[j0];
    rank += (o.x > mine) || (o.x == mine && (j0 + 0) < i);
    rank += (o.y > mine) || (o.y == mine && (j0 + 1) < i);
    rank += (o.z > mine) || (o.z == mine && (j0 + 2) < i);
    rank += (o.w > mine) || (o.w == mine && (j0 + 3) < i);
  }
  if (rank < NPROBE) top[rank] = i;
}

// ---------------- Phase 3+4: gathered GEMV via V_WMMA_F32_16X16X4_F32 ----------
// grid NCAND/16 = 512 blocks, block 256 (8 waves). Block handles 16 gathered
// rows; each wave covers a 512-wide K slice with 128 fp32 WMMAs.
// A[m,k] = W[row_m, kb+k]  (A layout: lanes 0-15 carry K=0,1; lanes 16-31 K=2,3)
// B[k,n] = h[kb+k] broadcast across n  -> every column of C is the full dot.
__global__ void __launch_bounds__(256)
k_logits(const float* __restrict__ h,
         const int*   __restrict__ top,
         const int*   __restrict__ vmap,
         const float* __restrict__ W,
         float* __restrict__ logits_out) {
  __shared__ float sh[D_DIM];
  __shared__ int   rows[16];
  __shared__ float part[8][16];

  const int tid = threadIdx.x;
  for (int i = tid * 4; i < D_DIM; i += 256 * 4)
    *(float4*)&sh[i] = *(const float4*)&h[i];
  if (tid < 16) {
    const int j  = blockIdx.x * 16 + tid;
    const int cl = top[j >> 6];
    rows[tid] = vmap[cl * CS + (j & (CS - 1))];
  }
  __syncthreads();

  const int wid  = tid >> 5;
  const int lane = tid & 31;
  const int m    = lane & 15;
  const int ksel = lane >> 4;                       // 0 -> k{0,1}, 1 -> k{2,3}
  const float* wrow = W + (size_t)rows[m] * D_DIM;

  v8f acc = {0.f, 0.f, 0.f, 0.f, 0.f, 0.f, 0.f, 0.f};
  const int k0 = wid * (D_DIM / 8);
  for (int kb0 = k0; kb0 < k0 + D_DIM / 8; kb0 += 32) {
    __builtin_prefetch(wrow + kb0 + 128, 0, 3);     // global_prefetch_b8, one line ahead
    #pragma unroll
    for (int kb = kb0; kb < kb0 + 32; kb += 4) {
      const v2f a = *(const v2f*)(wrow + kb + 2 * ksel);   // global_load_b64
      const v2f b = *(const v2f*)(&sh[kb + 2 * ksel]);     // ds_load_b64 (broadcast)
      acc = __builtin_amdgcn_wmma_f32_16x16x4_f32(
          /*neg_a=*/false, a, /*neg_b=*/false, b,
          /*c_mod=*/(short)0, acc, /*reuse_a=*/false, /*reuse_b=*/false);
    }
  }

  // C/D layout: lane 0 VGPR v = C[M=v, N=0]; lane 16 VGPR v = C[M=8+v, N=0].
  if (lane == 0) {
    #pragma unroll
    for (int v = 0; v < 8; ++v) part[wid][v] = acc[v];
  } else if (lane == 16) {
    #pragma unroll
    for (int v = 0; v < 8; ++v) part[wid][8 + v] = acc[v];
  }
  __syncthreads();
  if (tid < 16) {
    float sum = 0.f;
    #pragma unroll
    for (int w = 0; w < 8; ++w) sum += part[w][tid];
    logits_out[blockIdx.x * 16 + tid] = sum;
  }
}

// ---------------- Phase 5: lowest-index argmax -> vocab id ----------------
__global__ void k_argmax(const float* __restrict__ logits,
                         const int*   __restrict__ top,
                         const int*   __restrict__ vmap,
                         float* __restrict__ out0) {
  __shared__ float bv[256];
  __shared__ int   bi[256];
  const int tid = threadIdx.x;
  float best = -3.4e38f;
  int   bidx = NCAND;
  for (int j = tid; j < NCAND; j += 256) {          // strided: first hit = lowest idx
    const float v = logits[j];
    if (v > best) { best = v; bidx = j; }
  }
  bv[tid] = best; bi[tid] = bidx;
  __syncthreads();
  for (int s = 128; s > 0; s >>= 1) {
    if (tid < s) {
      if (bv[tid + s] > bv[tid] ||
          (bv[tid + s] == bv[tid] && bi[tid + s] < bi[tid])) {
        bv[tid] = bv[tid + s]; bi[tid] = bi[tid + s];
      }
    }
    __syncthreads();
  }
  if (tid == 0) {
    const int j  = bi[0];
    const int cl = top[j >> 6];
    out0[0] = (float)vmap[cl * CS + (j & (CS - 1))];
  }
}

extern "C" void kernel_launch(void* const* d_in, const int* in_sizes, int n_in,
                              void* d_out, int out_size, void* d_ws, size_t ws_size,
                              hipStream_t stream) {
  const float* h    = (const float*)d_in[0];   // [1,1,4096]
  const float* cent = (const float*)d_in[1];   // [4096,2048]
  const int*   vmap = (const int*)d_in[2];     // [2048,64]
  const float* W    = (const float*)d_in[3];   // [128000,4096]
  // d_in[4] = n_probes (=128, compile-time constant here)

  float* pdot = (float*)d_ws;                  // 16*2048 floats
  float* pnrm = pdot + 16 * NC;                // 16*2048 floats
  float* sims = pnrm + 16 * NC;                // 2048 floats
  int*   top  = (int*)(sims + NC);             // 128 ints

  float* out    = (float*)d_out;               // [0]=vocab id, [1..8192]=logits
  float* logits = out + 1;

  k_sims_partial<<<dim3(NC / 256, 16), 256, 0, stream>>>(h, cent, pdot, pnrm);
  k_sims_reduce <<<NC / 256, 256, 0, stream>>>(pdot, pnrm, sims);
  k_topk        <<<NC / 256, 256, 0, stream>>>(sims, top);
  k_logits      <<<NCAND / 16, 256, 0, stream>>>(h, top, vmap, W, logits);
  k_argmax      <<<1, 256, 0, stream>>>(logits, top, vmap, out);
}